// RotationPredHeadSim_64948495450586
// MI455X (gfx1250) — compile-verified
//
#include <hip/hip_runtime.h>
#include <hip/hip_bf16.h>
#include <math.h>

// ---------------------------------------------------------------------------
// RotationPredHeadSim for MI455X (gfx1250, wave32).
//
// Key simplifications proven from the reference math:
//  * angles are multiples of 90deg -> nearest-neighbor rotation is an exact
//    pixel permutation; rot(-a)(rot(a)(img)) == img; resize-to-same-shape is
//    identity.  So  p_c_nr == p_c  and  p_c_r[y,x] = p_c[src_t(y,x)].
//  * only the 8 `loc` channels of z_a / z_b are ever used.
//
// Bandwidth budget: ~100 MB of true traffic -> ~4.3 us at 23.3 TB/s.
// The X@W1 (16x4096x64 fp32) GEMM is mapped onto V_WMMA_F32_16X16X4_F32.
// ---------------------------------------------------------------------------

typedef float v2f __attribute__((ext_vector_type(2)));
typedef float v8f __attribute__((ext_vector_type(8)));

#define HW 256
#define PIX (HW * HW)          // 65536
#define NB 16                  // batch
#define NF 8                   // N_FEAT
#define NP 8                   // pool grid N
#define NCELL (NP * NP)        // 64
#define DIN 4096               // N**4
#define DH 64                  // hidden
#define DOUT 4                 // N_TEMP_ROTS

// workspace layout (floats)
#define WS_QNR 0                               // 16*65536
#define WS_QR (WS_QNR + NB * PIX)              // 16*65536
#define WS_ZAR (WS_QR + NB * PIX)              // 16*8*64
#define WS_ZBR (WS_ZAR + NB * NF * NCELL)      // 16*8*64
#define WS_X (WS_ZBR + NB * NF * NCELL)        // 16*4096
#define WS_H (WS_X + NB * DIN)                 // 16*64
// total = 2,180,096 floats = ~8.7 MB

// ---------------------------------------------------------------------------
// Kernel A: q_nr = 1 - sigmoid(o1-o0);  q_r = same, gathered through the
// exact 90-degree-rotation source permutation. 1 thread / pixel, coalesced
// writes; the gathered reads hit L2 (output is only 8 MB, L2 is 192 MB).
// ---------------------------------------------------------------------------
__global__ void mask_kernel(const float* __restrict__ output,
                            const int* __restrict__ target,
                            float* __restrict__ q_nr, float* __restrict__ q_r) {
  int idx = blockIdx.x * blockDim.x + threadIdx.x;   // over 16*65536
  int b = idx >> 16;
  int pix = idx & (PIX - 1);
  int y = pix >> 8, x = pix & 255;

  const float* ob = output + (size_t)b * 2 * PIX;
  // q_nr = 1 - softmax[:,1] = sigmoid(o0 - o1)
  float o0 = ob[pix];
  float o1 = ob[PIX + pix];
  q_nr[idx] = 1.0f / (1.0f + __expf(o1 - o0));

  // rotated-source gather
  int t = target[b] & 3;
  int sy, sx;
  switch (t) {
    case 0: sy = y;        sx = x;        break;
    case 1: sy = 255 - x;  sx = y;        break;  // rot 90
    case 2: sy = 255 - y;  sx = 255 - x;  break;  // rot 180
    default: sy = x;       sx = 255 - y;  break;  // rot 270
  }
  int sp = (sy << 8) + sx;
  float s0 = ob[sp];
  float s1 = ob[PIX + sp];
  q_r[idx] = 1.0f / (1.0f + __expf(s1 - s0));
}

// ---------------------------------------------------------------------------
// Kernel B: fused mask-multiply + 32x32 adaptive average pool over the 8
// selected channels.  One block = one (batch, feat, pool-row-band, side).
// 256 threads, thread t owns column t -> fully coalesced 1 KB row loads.
// Each wave32 covers exactly one 32-column pool cell -> __shfl_xor reduce.
// ---------------------------------------------------------------------------
__global__ void pool_kernel(const float* __restrict__ z_a,
                            const float* __restrict__ z_b,
                            const int* __restrict__ loc,
                            const float* __restrict__ q_nr,
                            const float* __restrict__ q_r,
                            float* __restrict__ z_ar, float* __restrict__ z_br) {
  int bi = blockIdx.x;                 // 16*8*8*2 = 2048 blocks
  int b = bi >> 7;
  int rem = bi & 127;
  int f = rem >> 4;
  int rem2 = rem & 15;
  int i = rem2 >> 1;                   // pool row 0..7
  int side = rem2 & 1;                 // 0 = z_a/q_nr, 1 = z_b/q_r

  int ch = loc[f];
  const float* zp = (side == 0 ? z_a : z_b) + ((size_t)b * 64 + ch) * PIX;
  const float* qp = (side == 0 ? q_nr : q_r) + (size_t)b * PIX;

  int x = threadIdx.x;                 // column
  float acc = 0.0f;
#pragma unroll 4
  for (int r = 0; r < 32; ++r) {
    int off = ((i * 32 + r) << 8) + x;
    acc += zp[off] * qp[off];
  }
  // reduce 32 columns (one wave) -> one pool cell
  for (int off = 16; off > 0; off >>= 1) acc += __shfl_xor(acc, off, 32);

  int lane = threadIdx.x & 31;
  int j = threadIdx.x >> 5;            // pool col 0..7
  if (lane == 0) {
    float* outp = (side == 0 ? z_ar : z_br);
    outp[((b * NF + f) * NCELL) + i * NP + j] = acc * (1.0f / 1024.0f);
  }
}

// ---------------------------------------------------------------------------
// Kernel C: per batch, C[n,m] = sum_k z_ar[k,n]*z_br[k,m] (64x64, K=8),
// softmax over n (columns of the flattened layout), write X[b, n*64+m].
// One block per batch; 64x64 C tile lives in LDS (16 KB of the 320 KB WGP).
// ---------------------------------------------------------------------------
__global__ void corr_softmax_kernel(const float* __restrict__ z_ar,
                                    const float* __restrict__ z_br,
                                    float* __restrict__ X) {
  __shared__ float Cs[NCELL * NCELL];      // 64x64
  __shared__ float za[NF * NCELL];
  __shared__ float zb[NF * NCELL];
  __shared__ float colmax[NCELL];
  __shared__ float colsum[NCELL];

  int b = blockIdx.x;
  int tid = threadIdx.x;                   // 256 threads

  for (int t = tid; t < NF * NCELL; t += 256) {
    za[t] = z_ar[b * NF * NCELL + t];
    zb[t] = z_br[b * NF * NCELL + t];
  }
  __syncthreads();

#pragma unroll
  for (int t = 0; t < 16; ++t) {
    int id = tid + 256 * t;                // id = n*64 + m
    int n = id >> 6, m = id & 63;
    float s = 0.0f;
#pragma unroll
    for (int k = 0; k < NF; ++k) s += za[k * NCELL + n] * zb[k * NCELL + m];
    Cs[id] = s;
  }
  __syncthreads();

  if (tid < NCELL) {                       // one thread per column m
    int m = tid;
    float mx = -1e30f;
    for (int n = 0; n < NCELL; ++n) mx = fmaxf(mx, Cs[n * NCELL + m]);
    float sum = 0.0f;
    for (int n = 0; n < NCELL; ++n) sum += __expf(Cs[n * NCELL + m] - mx);
    colmax[m] = mx;
    colsum[m] = 1.0f / sum;
  }
  __syncthreads();

#pragma unroll
  for (int t = 0; t < 16; ++t) {
    int id = tid + 256 * t;
    int m = id & 63;
    X[b * DIN + id] = __expf(Cs[id] - colmax[m]) * colsum[m];
  }
}

// ---------------------------------------------------------------------------
// Kernel D: H(16x64) = relu(X(16x4096) @ W1(4096x64) + b1) using
// V_WMMA_F32_16X16X4_F32.  One workgroup, 4 waves; wave w owns N-tile
// [16w,16w+16); M = the 16 batches; K-loop of 1024 WMMAs.
// ISA lane layouts (CDNA5 ISA):
//   A 16x4 f32 : lane l -> M = l&15 ; vgpr0 = K (+0 lanes0-15 / +2 lanes16-31),
//                vgpr1 = K+1 / K+3.
//   B 4x16 f32 : mirrored, N = lane&15.
//   D 16x16    : vgpr v -> M = v (+8 for lanes16-31), N = lane&15.
// ---------------------------------------------------------------------------
__global__ void mlp1_wmma_kernel(const float* __restrict__ X,
                                 const float* __restrict__ W1,
                                 const float* __restrict__ b1,
                                 float* __restrict__ H) {
  const int lane = threadIdx.x & 31;
  const int wave = threadIdx.x >> 5;       // 0..3
  const int m = lane & 15;                 // A row / B col within tile
  const int kh = (lane < 16) ? 0 : 2;      // K half select
  const int n = wave * 16 + m;             // global hidden index

  v8f acc = {0.f, 0.f, 0.f, 0.f, 0.f, 0.f, 0.f, 0.f};
  const float* xrow = X + m * DIN;

  for (int k = 0; k < DIN; k += 4) {
    v2f a, bm;
    a.x = xrow[k + kh];
    a.y = xrow[k + kh + 1];
    bm.x = W1[(size_t)(k + kh) * DH + n];
    bm.y = W1[(size_t)(k + kh + 1) * DH + n];
    acc = __builtin_amdgcn_wmma_f32_16x16x4_f32(
        /*neg_a=*/false, a, /*neg_b=*/false, bm,
        /*c_mod=*/(short)0, acc, /*reuse_a=*/false, /*reuse_b=*/false);
  }

  float bias = b1[n];
#pragma unroll
  for (int v = 0; v < 8; ++v) {
    int row = v + ((lane < 16) ? 0 : 8);   // batch index
    float h = acc[v] + bias;
    H[row * DH + n] = fmaxf(h, 0.0f);
  }
}

// ---------------------------------------------------------------------------
// Kernel E: r(16x4) = H @ W2 + b2.  64 outputs, trivial.
// ---------------------------------------------------------------------------
__global__ void mlp2_kernel(const float* __restrict__ H,
                            const float* __restrict__ W2,
                            const float* __restrict__ b2,
                            float* __restrict__ out) {
  int tid = threadIdx.x;                   // 64 threads
  int b = tid >> 2, j = tid & 3;
  float s = b2[j];
#pragma unroll 8
  for (int k = 0; k < DH; ++k) s += H[b * DH + k] * W2[k * DOUT + j];
  out[b * DOUT + j] = s;
}

// ---------------------------------------------------------------------------
extern "C" void kernel_launch(void* const* d_in, const int* in_sizes, int n_in,
                              void* d_out, int out_size, void* d_ws, size_t ws_size,
                              hipStream_t stream) {
  const float* z_a    = (const float*)d_in[0];
  const float* z_b    = (const float*)d_in[1];
  const float* output = (const float*)d_in[2];
  const int*   target = (const int*)d_in[3];
  const int*   loc    = (const int*)d_in[4];
  const float* W1     = (const float*)d_in[5];
  const float* b1     = (const float*)d_in[6];
  const float* W2     = (const float*)d_in[7];
  const float* b2     = (const float*)d_in[8];

  float* ws   = (float*)d_ws;
  float* q_nr = ws + WS_QNR;
  float* q_r  = ws + WS_QR;
  float* z_ar = ws + WS_ZAR;
  float* z_br = ws + WS_ZBR;
  float* X    = ws + WS_X;
  float* H    = ws + WS_H;

  // A: per-pixel masks (16*65536 pixels)
  mask_kernel<<<(NB * PIX) / 256, 256, 0, stream>>>(output, target, q_nr, q_r);

  // B: fused mask * z + 32x32 average pool (2048 blocks)
  pool_kernel<<<NB * NF * NP * 2, 256, 0, stream>>>(z_a, z_b, loc, q_nr, q_r,
                                                    z_ar, z_br);

  // C: 64x64 correlation + column softmax, per batch
  corr_softmax_kernel<<<NB, 256, 0, stream>>>(z_ar, z_br, X);

  // D: X @ W1 + b1, relu — fp32 WMMA
  mlp1_wmma_kernel<<<1, 128, 0, stream>>>(X, W1, b1, H);

  // E: H @ W2 + b2 -> d_out (16x4)
  mlp2_kernel<<<1, 64, 0, stream>>>(H, W2, b2, (float*)d_out);
}